// SlotAttention_24103356465702
// MI455X (gfx1250) — compile-verified
//
#include <hip/hip_runtime.h>
#include <math.h>

typedef __attribute__((ext_vector_type(2))) float v2f;
typedef __attribute__((ext_vector_type(8))) float v8f;

#define Dd 128
#define Nn 4096
#define Bb 32
#define Kk 8
#define NITERS 3
#define ATT_SCALE 0.08838834764831845f /* 128^-0.5 */
#define LN_EPS 1e-5f

// ---------------------------------------------------------------------------
// WMMA helpers: f32 16x16x4, D = A(16xK) * op(B) + C
// A-frag layout: lane L -> row m=L%16, k-pair base kh = 2*(L/16)
// B-frag layout: lane L -> col n=L%16, same k-pair base
// C/D layout:    lane L -> col n=L%16, rows (L/16)*8 + i  (i = vgpr index)
// ---------------------------------------------------------------------------

// out[m][n] = sum_k A[m][k] * W[nbase+n][k]   (A row-major, W row-major [N][K])
__device__ __forceinline__ v8f wmma_axwt(const float* __restrict__ A, int lda,
                                         const float* __restrict__ W, int ldw,
                                         int kdim, v8f c) {
  const int lane = threadIdx.x & 31;
  const int r = lane & 15;
  const int kh = (lane >> 4) << 1;
  const float* arow = A + r * lda + kh;
  const float* wrow = W + r * ldw + kh;
#pragma unroll 8
  for (int k = 0; k < kdim; k += 4) {
    v2f a, b;
    a.x = arow[k];     a.y = arow[k + 1];
    b.x = wrow[k];     b.y = wrow[k + 1];
    c = __builtin_amdgcn_wmma_f32_16x16x4_f32(false, a, false, b, (short)0, c,
                                              false, false);
  }
  return c;
}

// store 16x16 C tile to O (row-major, leading dim ldo)
__device__ __forceinline__ void store_tile(float* __restrict__ O, int ldo, v8f c) {
  const int lane = threadIdx.x & 31;
  const int n = lane & 15;
  const int mb = (lane >> 4) << 3;
#pragma unroll
  for (int i = 0; i < 8; ++i) O[(mb + i) * ldo + n] = c[i];
}

// layernorm of one row of Dd elements (dst may alias src)
__device__ __forceinline__ void ln_row(const float* __restrict__ src,
                                       float* __restrict__ dst,
                                       const float* __restrict__ g,
                                       const float* __restrict__ be) {
  float m = 0.f;
  for (int c = 0; c < Dd; ++c) m += src[c];
  m *= (1.0f / Dd);
  float var = 0.f;
  for (int c = 0; c < Dd; ++c) { float d = src[c] - m; var += d * d; }
  var *= (1.0f / Dd);
  const float inv = rsqrtf(var + LN_EPS);
  for (int c = 0; c < Dd; ++c) dst[c] = (src[c] - m) * inv * g[c] + be[c];
}

// ---------------------------------------------------------------------------
// slots = mu + exp(logsigma) * noise
// ---------------------------------------------------------------------------
__global__ void slots_init_kernel(const float* __restrict__ mu,
                                  const float* __restrict__ logsig,
                                  const float* __restrict__ noise,
                                  float* __restrict__ slots) {
  int i = blockIdx.x * blockDim.x + threadIdx.x;
  if (i < Bb * Kk * Dd) {
    int kd = i % (Kk * Dd);
    slots[i] = mu[kd] + __expf(logsig[kd]) * noise[i];
  }
}

// ---------------------------------------------------------------------------
// xn = LN(x); k = xn@Wk^T + bk; v = xn@Wv^T + bv.  One wave per 16 rows.
// ---------------------------------------------------------------------------
__global__ void __launch_bounds__(32)
kv_proj_kernel(const float* __restrict__ x, const float* __restrict__ g_in,
               const float* __restrict__ be_in, const float* __restrict__ Wk,
               const float* __restrict__ bk, const float* __restrict__ Wv,
               const float* __restrict__ bv, float* __restrict__ kout,
               float* __restrict__ vout) {
  __shared__ float xt[16 * Dd];
  const int rowbase = blockIdx.x * 16;
  const int tid = threadIdx.x;
  __builtin_prefetch(Wk + (tid << 5), 0, 3);
  __builtin_prefetch(Wv + (tid << 5), 0, 3);
  for (int i = tid; i < 16 * Dd; i += 32) xt[i] = x[(size_t)rowbase * Dd + i];
  __syncthreads();
  if (tid < 16) ln_row(xt + tid * Dd, xt + tid * Dd, g_in, be_in);
  __syncthreads();
  const int n = tid & 15;
  for (int t = 0; t < 8; ++t) {
    const int nb = t * 16;
    v8f c = {};
    c = wmma_axwt(xt, Dd, Wk + nb * Dd, Dd, Dd, c);
    float bias = bk[nb + n];
#pragma unroll
    for (int i = 0; i < 8; ++i) c[i] += bias;
    store_tile(kout + (size_t)rowbase * Dd + nb, Dd, c);
    v8f c2 = {};
    c2 = wmma_axwt(xt, Dd, Wv + nb * Dd, Dd, Dd, c2);
    bias = bv[nb + n];
#pragma unroll
    for (int i = 0; i < 8; ++i) c2[i] += bias;
    store_tile(vout + (size_t)rowbase * Dd + nb, Dd, c2);
  }
}

// ---------------------------------------------------------------------------
// q = LN(slots)@Wq^T + bq; also zero rowsum for this iteration.
// ---------------------------------------------------------------------------
__global__ void __launch_bounds__(32)
q_proj_kernel(const float* __restrict__ slots, const float* __restrict__ g_sl,
              const float* __restrict__ be_sl, const float* __restrict__ Wq,
              const float* __restrict__ bq, float* __restrict__ q,
              float* __restrict__ rowsum) {
  __shared__ float st[16 * Dd];
  const int rowbase = blockIdx.x * 16;
  const int tid = threadIdx.x;
  for (int i = tid; i < 16 * Dd; i += 32) st[i] = slots[rowbase * Dd + i];
  if (tid < 16) rowsum[blockIdx.x * 16 + tid] = 0.f;
  __syncthreads();
  if (tid < 16) ln_row(st + tid * Dd, st + tid * Dd, g_sl, be_sl);
  __syncthreads();
  const int n = tid & 15;
  for (int t = 0; t < 8; ++t) {
    const int nb = t * 16;
    v8f c = {};
    c = wmma_axwt(st, Dd, Wq + nb * Dd, Dd, Dd, c);
    const float bias = bq[nb + n];
#pragma unroll
    for (int i = 0; i < 8; ++i) c[i] += bias;
    store_tile(q + rowbase * Dd + nb, Dd, c);
  }
}

// ---------------------------------------------------------------------------
// logits = q@k^T * SCALE; softmax over 8 slots (in-lane!); +1e-8;
// store unnormalized attn and accumulate per-slot row sums.
// Block: 8 waves, each wave covers 16 input columns j.
// ---------------------------------------------------------------------------
__global__ void __launch_bounds__(256)
attn_logits_kernel(const float* __restrict__ q, const float* __restrict__ kmat,
                   float* __restrict__ attn, float* __restrict__ rowsum) {
  __shared__ float qt[16 * Dd];
  __shared__ float bsum[Kk];
  const int b = blockIdx.x >> 5;
  const int chunk = blockIdx.x & 31;
  const int tid = threadIdx.x;
  for (int i = tid; i < 16 * Dd; i += 256) {
    int r = i / Dd;
    qt[i] = (r < Kk) ? q[(b * Kk + r) * Dd + (i % Dd)] : 0.f;
  }
  if (tid < Kk) bsum[tid] = 0.f;
  __syncthreads();
  const int wave = tid >> 5, lane = tid & 31;
  const int jbase = chunk * 128 + wave * 16;
  v8f c = {};
  c = wmma_axwt(qt, Dd, kmat + (size_t)(b * Nn + jbase) * Dd, Dd, Dd, c);
  if (lane < 16) {  // lanes 0-15 hold all 8 slot logits for column j in c[0..7]
    const int j = jbase + lane;
    float mx = -1e30f;
#pragma unroll
    for (int i = 0; i < Kk; ++i) { c[i] *= ATT_SCALE; mx = fmaxf(mx, c[i]); }
    float s = 0.f;
#pragma unroll
    for (int i = 0; i < Kk; ++i) { float e = __expf(c[i] - mx); c[i] = e; s += e; }
    const float inv = 1.0f / s;
#pragma unroll
    for (int i = 0; i < Kk; ++i) {
      float p = c[i] * inv + 1e-8f;
      attn[(size_t)(b * Kk + i) * Nn + j] = p;
      atomicAdd(&bsum[i], p);
    }
  }
  __syncthreads();
  if (tid < Kk) atomicAdd(&rowsum[b * Kk + tid], bsum[tid]);
}

// ---------------------------------------------------------------------------
// updates[b,i,d] = (sum_j attn[b,i,j] * v[b,j,d]) / rowsum[b,i]
// One block per batch; wave w owns d-tile [16w,16w+16); K loop over N=4096.
// ---------------------------------------------------------------------------
__global__ void __launch_bounds__(256)
updates_kernel(const float* __restrict__ attn, const float* __restrict__ rowsum,
               const float* __restrict__ v, float* __restrict__ updates) {
  const int b = blockIdx.x;
  const int tid = threadIdx.x;
  const int wave = tid >> 5, lane = tid & 31;
  const int r = lane & 15;
  const int kh = (lane >> 4) << 1;
  const int dbase = wave * 16;
  const float amask = (r < Kk) ? 1.f : 0.f;  // pad slot rows 8..15 with zeros
  const float* arow = attn + (size_t)(b * Kk + (r & 7)) * Nn + kh;
  const float* bcol = v + (size_t)(b * Nn + kh) * Dd + dbase + r;
  v8f c = {};
#pragma unroll 4
  for (int k = 0; k < Nn; k += 4) {
    v2f a, bb;
    a.x = arow[k] * amask;          a.y = arow[k + 1] * amask;
    bb.x = bcol[(size_t)k * Dd];    bb.y = bcol[(size_t)(k + 1) * Dd];
    c = __builtin_amdgcn_wmma_f32_16x16x4_f32(false, a, false, bb, (short)0, c,
                                              false, false);
  }
  if (lane < 16) {  // rows 0..7 live in lanes 0..15
#pragma unroll
    for (int i = 0; i < Kk; ++i) {
      const float invr = 1.0f / rowsum[b * Kk + i];
      updates[(size_t)(b * Kk + i) * Dd + dbase + r] = c[i] * invr;
    }
  }
}

// ---------------------------------------------------------------------------
// GRU cell + LN + FFN residual. One block per 16 rows of the 256 slot rows.
// Shared memory (64KB) with buffer aliasing: u->ns, h->ff, gi->hid.
// ---------------------------------------------------------------------------
__global__ void __launch_bounds__(256)
gru_ffn_kernel(const float* __restrict__ updates, float* __restrict__ slots,
               const float* __restrict__ W_ih, const float* __restrict__ b_ih,
               const float* __restrict__ W_hh, const float* __restrict__ b_hh,
               const float* __restrict__ g_ff, const float* __restrict__ be_ff,
               const float* __restrict__ W1, const float* __restrict__ b1,
               const float* __restrict__ W2, const float* __restrict__ b2) {
  __shared__ float smem[16384];  // 64 KB
  float* u = smem;               // 16x128
  float* h = smem + 2048;        // 16x128
  float* gi = smem + 4096;       // 16x384
  float* gh = smem + 10240;      // 16x384
  float* ns = u;                 // reuse after GEMM phase
  float* ff = h;                 // reuse after gate phase
  float* hid = gi;               // reuse after gate phase

  const int rowbase = blockIdx.x * 16;
  const int tid = threadIdx.x;
  for (int i = tid; i < 16 * Dd; i += 256) {
    u[i] = updates[rowbase * Dd + i];
    h[i] = slots[rowbase * Dd + i];
  }
  __syncthreads();
  const int wave = tid >> 5, lane = tid & 31;
  const int n = lane & 15;
  // 24 gi tiles + 24 gh tiles (N=384 -> 24 tiles each), 6 per wave
  for (int s = 0; s < 6; ++s) {
    const int t = wave * 6 + s;
    const bool isgh = t >= 24;
    const int nt = isgh ? t - 24 : t;
    const float* A = isgh ? h : u;
    const float* W = isgh ? W_hh : W_ih;
    const float* bias = isgh ? b_hh : b_ih;
    float* Ot = isgh ? gh : gi;
    v8f c = {};
    c = wmma_axwt(A, Dd, W + nt * 16 * Dd, Dd, Dd, c);
    const float bv2 = bias[nt * 16 + n];
#pragma unroll
    for (int i = 0; i < 8; ++i) c[i] += bv2;
    store_tile(Ot + nt * 16, 3 * Dd, c);
  }
  __syncthreads();
  // gates (torch order r,z,n): new = (1-z)*tanh(i_n + r*h_n) + z*h
  for (int e = tid; e < 16 * Dd; e += 256) {
    const int row = e / Dd, col = e % Dd;
    const float* gir = gi + row * 3 * Dd;
    const float* ghr = gh + row * 3 * Dd;
    const float r_ = 1.f / (1.f + __expf(-(gir[col] + ghr[col])));
    const float z_ = 1.f / (1.f + __expf(-(gir[Dd + col] + ghr[Dd + col])));
    const float nn_ = tanhf(gir[2 * Dd + col] + r_ * ghr[2 * Dd + col]);
    const float hval = h[e];
    __syncthreads();  // ensure all gate reads of h done before ff overwrites it
    ns[e] = (1.f - z_) * nn_ + z_ * hval;
  }
  __syncthreads();
  if (tid < 16) ln_row(ns + tid * Dd, ff + tid * Dd, g_ff, be_ff);
  __syncthreads();
  {  // hid = relu(ff @ W1^T + b1); wave w -> tile w
    const int nb = wave * 16;
    v8f c = {};
    c = wmma_axwt(ff, Dd, W1 + nb * Dd, Dd, Dd, c);
    const float bb = b1[nb + n];
#pragma unroll
    for (int i = 0; i < 8; ++i) c[i] = fmaxf(c[i] + bb, 0.f);
    __syncthreads();  // all reads of ff done before hid (aliases gi) written
    store_tile(hid + nb, Dd, c);
  }
  __syncthreads();
  {  // slots = ns + hid @ W2^T + b2
    const int nb = wave * 16;
    v8f c = {};
    c = wmma_axwt(hid, Dd, W2 + nb * Dd, Dd, Dd, c);
    const float bb = b2[nb + n];
    const int mb = (lane >> 4) << 3;
#pragma unroll
    for (int i = 0; i < 8; ++i) {
      const int row = mb + i;
      slots[(rowbase + row) * Dd + nb + n] = ns[row * Dd + nb + n] + c[i] + bb;
    }
  }
}

// ---------------------------------------------------------------------------
// out = [slots (B*K*D) | attn^T (B*N*K)], attn normalized by rowsum.
// ---------------------------------------------------------------------------
__global__ void finalize_kernel(const float* __restrict__ slots,
                                const float* __restrict__ attn,
                                const float* __restrict__ rowsum,
                                float* __restrict__ out) {
  const int t = blockIdx.x * blockDim.x + threadIdx.x;
  if (t < Bb * Kk * Dd) out[t] = slots[t];
  if (t < Bb * Nn) {
    const int b = t / Nn, j = t % Nn;
    float* o = out + Bb * Kk * Dd + (size_t)t * Kk;
#pragma unroll
    for (int i = 0; i < Kk; ++i) {
      const float p = attn[(size_t)(b * Kk + i) * Nn + j];
      o[i] = p / rowsum[b * Kk + i];
    }
  }
}

// ---------------------------------------------------------------------------
extern "C" void kernel_launch(void* const* d_in, const int* in_sizes, int n_in,
                              void* d_out, int out_size, void* d_ws,
                              size_t ws_size, hipStream_t stream) {
  (void)in_sizes; (void)n_in; (void)out_size; (void)ws_size;
  const float* x      = (const float*)d_in[0];
  const float* noise  = (const float*)d_in[1];
  const float* mu     = (const float*)d_in[2];
  const float* logsig = (const float*)d_in[3];
  const float* Wq     = (const float*)d_in[4];
  const float* bq     = (const float*)d_in[5];
  const float* Wk     = (const float*)d_in[6];
  const float* bk     = (const float*)d_in[7];
  const float* Wv     = (const float*)d_in[8];
  const float* bv     = (const float*)d_in[9];
  const float* W_ih   = (const float*)d_in[10];
  const float* b_ih   = (const float*)d_in[11];
  const float* W_hh   = (const float*)d_in[12];
  const float* b_hh   = (const float*)d_in[13];
  const float* W1     = (const float*)d_in[14];
  const float* b1     = (const float*)d_in[15];
  const float* W2     = (const float*)d_in[16];
  const float* b2     = (const float*)d_in[17];
  const float* g_in   = (const float*)d_in[18];
  const float* be_in  = (const float*)d_in[19];
  const float* g_sl   = (const float*)d_in[20];
  const float* be_sl  = (const float*)d_in[21];
  const float* g_ff   = (const float*)d_in[22];
  const float* be_ff  = (const float*)d_in[23];

  float* ws = (float*)d_ws;
  float* kbuf   = ws;                       // B*N*D = 16777216
  float* vbuf   = kbuf + (size_t)Bb * Nn * Dd;
  float* qbuf   = vbuf + (size_t)Bb * Nn * Dd;     // B*K*D = 32768
  float* attnb  = qbuf + (size_t)Bb * Kk * Dd;     // B*K*N = 1048576
  float* rowsum = attnb + (size_t)Bb * Kk * Nn;    // B*K   = 256
  float* slots  = rowsum + Bb * Kk;                // B*K*D
  float* upd    = slots + (size_t)Bb * Kk * Dd;    // B*K*D

  slots_init_kernel<<<(Bb * Kk * Dd + 255) / 256, 256, 0, stream>>>(
      mu, logsig, noise, slots);
  kv_proj_kernel<<<(Bb * Nn) / 16, 32, 0, stream>>>(x, g_in, be_in, Wk, bk, Wv,
                                                    bv, kbuf, vbuf);
  for (int it = 0; it < NITERS; ++it) {
    q_proj_kernel<<<(Bb * Kk) / 16, 32, 0, stream>>>(slots, g_sl, be_sl, Wq, bq,
                                                     qbuf, rowsum);
    attn_logits_kernel<<<Bb * (Nn / 128), 256, 0, stream>>>(qbuf, kbuf, attnb,
                                                            rowsum);
    updates_kernel<<<Bb, 256, 0, stream>>>(attnb, rowsum, vbuf, upd);
    gru_ffn_kernel<<<(Bb * Kk) / 16, 256, 0, stream>>>(
        upd, slots, W_ih, b_ih, W_hh, b_hh, g_ff, be_ff, W1, b1, W2, b2);
  }
  finalize_kernel<<<(Bb * Nn + 255) / 256, 256, 0, stream>>>(slots, attnb,
                                                             rowsum,
                                                             (float*)d_out);
}